// PM_MoE_Att_Block_53824530153848
// MI455X (gfx1250) — compile-verified
//
#include <hip/hip_runtime.h>
#include <hip/hip_bf16.h>
#include <math.h>

// dims from the reference
#define NB 512      // B
#define KS 2        // K routed slots
#define SN 128      // S
#define DN 128      // D
#define EN 4        // E experts
#define NH 8        // heads
#define FN 512      // ffn dim
#define HD 16       // head dim
#define WAVES 8
#define NTHR 256
#define MAXN_C 0.99999f  // 1 - 1e-5

typedef __attribute__((ext_vector_type(2))) float v2f;
typedef __attribute__((ext_vector_type(8))) float v8f;

__device__ __forceinline__ float wsum(float v) {
  v += __shfl_xor(v, 16); v += __shfl_xor(v, 8); v += __shfl_xor(v, 4);
  v += __shfl_xor(v, 2);  v += __shfl_xor(v, 1);
  return v;
}
__device__ __forceinline__ float wmaxr(float v) {
  v = fmaxf(v, __shfl_xor(v, 16)); v = fmaxf(v, __shfl_xor(v, 8));
  v = fmaxf(v, __shfl_xor(v, 4));  v = fmaxf(v, __shfl_xor(v, 2));
  v = fmaxf(v, __shfl_xor(v, 1));
  return v;
}

// C[m-tile mt, nt=0..NT-1] += A[128 x Kd] * B[Kd x NT*16]
// BT=true : Bsrc is "weight" layout, Bsrc[n*ldb + k]  (i.e. x @ W^T)
// BT=false: Bsrc is row-major B[k*ldb + n]
// Fragment layouts per CDNA5 ISA 7.12.2 (f32 16x16x4, wave32).
template<int NT, bool BT>
__device__ __forceinline__ void gemm_acc(const float* __restrict__ A, int lda,
                                         const float* __restrict__ Bsrc, int ldb,
                                         int Kd, v8f* acc, int mt, int lane) {
  const int half = lane >> 4, l16 = lane & 15;
  const float* arow = A + (mt * 16 + l16) * lda + 2 * half;
  for (int k0 = 0; k0 < Kd; k0 += 4) {
    v2f a = *(const v2f*)(arow + k0);
#pragma unroll
    for (int nt = 0; nt < NT; ++nt) {
      v2f bf;
      if (BT) {
        bf = *(const v2f*)(Bsrc + (nt * 16 + l16) * ldb + k0 + 2 * half);
      } else {
        bf.x = Bsrc[(k0 + 2 * half) * ldb + nt * 16 + l16];
        bf.y = Bsrc[(k0 + 2 * half + 1) * ldb + nt * 16 + l16];
      }
      acc[nt] = __builtin_amdgcn_wmma_f32_16x16x4_f32(false, a, false, bf,
                                                      (short)0, acc[nt], false, false);
    }
  }
}

template<int NT>
__device__ __forceinline__ void store_tiles(float* dst, int ldc, const v8f* acc,
                                            int mt, int lane) {
  const int half = lane >> 4, l16 = lane & 15;
#pragma unroll
  for (int nt = 0; nt < NT; ++nt)
#pragma unroll
    for (int i = 0; i < 8; ++i)
      dst[(mt * 16 + half * 8 + i) * ldc + nt * 16 + l16] = acc[nt][i];
}

// dst = clipnorm( layernorm( logmap0(src) ), LCLIP )   (one wave per row)
__device__ __forceinline__ void rowwise_ln_clip(const float* __restrict__ src, int sld,
                                                const float* __restrict__ g,
                                                const float* __restrict__ bv,
                                                float* dst, int wid, int lane, float LCLIP) {
  for (int r = wid; r < SN; r += WAVES) {
    float xv[4]; float ss = 0.f;
#pragma unroll
    for (int j = 0; j < 4; ++j) { xv[j] = src[r * sld + lane + 32 * j]; ss += xv[j] * xv[j]; }
    ss = wsum(ss);
    float n = sqrtf(fmaxf(ss, 1e-15f));
    float ls = atanhf(fminf(n, MAXN_C)) / n;          // logmap0 scale
    float u[4]; float m = 0.f;
#pragma unroll
    for (int j = 0; j < 4; ++j) { u[j] = ls * xv[j]; m += u[j]; }
    m = wsum(m) * (1.f / (float)DN);
    float var = 0.f;
#pragma unroll
    for (int j = 0; j < 4; ++j) { float d = u[j] - m; var += d * d; }
    var = wsum(var) * (1.f / (float)DN);
    float inv = rsqrtf(var + 1e-5f);
    float ln[4]; float s2 = 0.f;
#pragma unroll
    for (int j = 0; j < 4; ++j) {
      int c = lane + 32 * j;
      ln[j] = g[c] * (u[j] - m) * inv + bv[c];
      s2 += ln[j] * ln[j];
    }
    s2 = wsum(s2);
    float nl = sqrtf(fmaxf(s2, 1e-15f));
    float cs = fminf(nl, LCLIP) / nl;                  // logmap0(expmap0(.)) == norm clip
#pragma unroll
    for (int j = 0; j < 4; ++j) dst[r * DN + lane + 32 * j] = cs * ln[j];
  }
}

// In-place: Z_row <- logmap0( projx( mobius_add( expmap0(Z_row), ytil ) ) )
// All tail scalars are analytic functions of (||z||^2, <z,ytil>).
__device__ __forceinline__ void rowwise_mantail_log(float* Z, const float* __restrict__ yt,
                                                    float y2, int wid, int lane) {
  for (int r = wid; r < SN; r += WAVES) {
    float z[4]; float sz = 0.f, szy = 0.f;
#pragma unroll
    for (int j = 0; j < 4; ++j) {
      int c = lane + 32 * j;
      z[j] = Z[r * DN + c];
      sz += z[j] * z[j]; szy += z[j] * yt[c];
    }
    sz = wsum(sz); szy = wsum(szy);
    float zn = sqrtf(fmaxf(sz, 1e-15f));
    float th = tanhf(zn), s1 = th / zn;
    float xy = s1 * szy, x2 = th * th;
    float A1 = 1.f + 2.f * xy + y2, B1 = 1.f - x2;
    float den = fmaxf(1.f + 2.f * xy + x2 * y2, 1e-15f);
    float nm = sqrtf(fmaxf(A1 * A1 * x2 + 2.f * A1 * B1 * xy + B1 * B1 * y2, 1e-30f)) / den;
    float np = fminf(nm, MAXN_C);
    float f = atanhf(np) / nm;                         // projx + logmap0 fused
    float al = f * A1 * s1 / den, be = f * B1 / den;
#pragma unroll
    for (int j = 0; j < 4; ++j) {
      int c = lane + 32 * j;
      Z[r * DN + c] = al * z[j] + be * yt[c];
    }
  }
}

// In-place norm clip of each row to CL (logmap0(expmap0(.)))
__device__ __forceinline__ void rowwise_clip(float* buf, float CL, int wid, int lane) {
  for (int r = wid; r < SN; r += WAVES) {
    float v[4]; float ss = 0.f;
#pragma unroll
    for (int j = 0; j < 4; ++j) { v[j] = buf[r * DN + lane + 32 * j]; ss += v[j] * v[j]; }
    ss = wsum(ss);
    float n = sqrtf(fmaxf(ss, 1e-15f));
    float sc = fminf(n, CL) / n;
#pragma unroll
    for (int j = 0; j < 4; ++j) buf[r * DN + lane + 32 * j] = sc * v[j];
  }
}

__global__ __launch_bounds__(NTHR)
void PM_MoE_Att_Block_53824530153848_kernel(
    const float* __restrict__ feat, const int* __restrict__ eidx,
    const unsigned char* __restrict__ pmask,
    const float* __restrict__ l1g, const float* __restrict__ l1b,
    const float* __restrict__ l2g, const float* __restrict__ l2b,
    const float* __restrict__ pwq, const float* __restrict__ pbq,
    const float* __restrict__ pwk, const float* __restrict__ pbk,
    const float* __restrict__ pwv, const float* __restrict__ pbv,
    const float* __restrict__ pwo, const float* __restrict__ pbo,
    const float* __restrict__ pf1w, const float* __restrict__ pf1b,
    const float* __restrict__ pf2w, const float* __restrict__ pf2b,
    float* __restrict__ out) {
  extern __shared__ float smem[];
  float* Hbuf = smem;                 // 16384 (A-matrix / scores / Zo)
  float* Qbuf = Hbuf + SN * DN;       // 16384 (q / ctx / fc1-chunk)
  float* Kbuf = Qbuf + SN * DN;       // 16384 (k / z2)
  float* Vbuf = Kbuf + SN * DN;       // 16384 (v / residual x2)
  float* yq   = Vbuf + SN * DN;       // 128 expmap0(bq)
  float* yk   = yq + DN;              // 128
  float* yv   = yk + DN;              // 128
  float* yo   = yv + DN;              // 128
  float* y2v  = yo + DN;              // 128 expmap0(fc2_b)
  float* y1v  = y2v + DN;             // 512 expmap0(fc1_b)
  float* padb = y1v + FN;             // 128 pad bias
  float* z1sq = padb + SN;            // 128
  float* z1y  = z1sq + SN;            // 128
  float* alA  = z1y + SN;             // 128
  float* beA  = alA + SN;             // 128
  float* rlsq = beA + SN;             // 128
  float* y2s  = rlsq + SN;            // 8 scalars (||expmap0(bias)||^2)

  const int blk = blockIdx.x;
  const int b = blk / EN, e = blk % EN;
  const int tid = threadIdx.x, wid = tid >> 5, lane = tid & 31;
  float* out_be = out + (size_t)(b * EN + e) * (SN * DN);

  int slot = -1;
  if (eidx[b * KS + 0] == e) slot = 0;
  else if (eidx[b * KS + 1] == e) slot = 1;
  if (slot < 0) {  // expert not routed: zero output tile
    for (int i = tid; i < SN * DN; i += NTHR) out_be[i] = 0.f;
    return;
  }

  const float LCLIP = atanhf(MAXN_C);
  const float* xg = feat + (size_t)(b * KS + slot) * (SN * DN);
  const float* Wq = pwq + e * DN * DN; const float* Wk = pwk + e * DN * DN;
  const float* Wv = pwv + e * DN * DN; const float* Wo = pwo + e * DN * DN;
  const float* F1 = pf1w + e * FN * DN; const float* F2 = pf2w + e * DN * FN;

  // ---- phase 0: pad bias + bias expmaps (one wave per bias vector) ----
  for (int s = tid; s < SN; s += NTHR) padb[s] = pmask[b * SN + s] ? -1e9f : 0.f;
  if (wid < 6) {
    const float* src; float* dst; int L = DN;
    switch (wid) {
      case 0: src = pbq + e * DN;  dst = yq;  break;
      case 1: src = pbk + e * DN;  dst = yk;  break;
      case 2: src = pbv + e * DN;  dst = yv;  break;
      case 3: src = pbo + e * DN;  dst = yo;  break;
      case 4: src = pf2b + e * DN; dst = y2v; break;
      default: src = pf1b + e * FN; dst = y1v; L = FN; break;
    }
    float ss = 0.f;
    for (int i = lane; i < L; i += 32) { float v = src[i]; ss += v * v; }
    ss = wsum(ss);
    float n = sqrtf(fmaxf(ss, 1e-15f));
    float th = tanhf(n), sc = th / n;
    for (int i = lane; i < L; i += 32) dst[i] = sc * src[i];
    if (lane == 0) y2s[wid] = th * th;
  }
  __syncthreads();

  // ---- phase 1: H = clip(LN1(logmap0(x))) ----
  rowwise_ln_clip(xg, DN, l1g + e * DN, l1b + e * DN, Hbuf, wid, lane, LCLIP);
  __syncthreads();

  // ---- phase 2: Q/K/V projections (WMMA) + Möbius tails ----
  { v8f acc[8] = {}; gemm_acc<8, true>(Hbuf, DN, Wq, DN, DN, acc, wid, lane);
    store_tiles<8>(Qbuf, DN, acc, wid, lane); }
  { v8f acc[8] = {}; gemm_acc<8, true>(Hbuf, DN, Wk, DN, DN, acc, wid, lane);
    store_tiles<8>(Kbuf, DN, acc, wid, lane); }
  { v8f acc[8] = {}; gemm_acc<8, true>(Hbuf, DN, Wv, DN, DN, acc, wid, lane);
    store_tiles<8>(Vbuf, DN, acc, wid, lane); }
  __syncthreads();
  rowwise_mantail_log(Qbuf, yq, y2s[0], wid, lane);
  rowwise_mantail_log(Kbuf, yk, y2s[1], wid, lane);
  rowwise_mantail_log(Vbuf, yv, y2s[2], wid, lane);
  __syncthreads();

  // ---- phase 3: attention, head by head (scores reuse Hbuf) ----
  float* SC = Hbuf;
  for (int h = 0; h < NH; ++h) {
    { v8f acc[8] = {};
      gemm_acc<8, true>(Qbuf + h * HD, DN, Kbuf + h * HD, DN, HD, acc, wid, lane);
      const int half = lane >> 4, l16 = lane & 15;
#pragma unroll
      for (int nt = 0; nt < 8; ++nt)
#pragma unroll
        for (int i = 0; i < 8; ++i)
          SC[(wid * 16 + half * 8 + i) * SN + nt * 16 + l16] =
              acc[nt][i] * 0.25f + padb[nt * 16 + l16];
    }
    __syncthreads();
    for (int r = wid; r < SN; r += WAVES) {  // softmax rows
      float vv[4]; float mx = -3.0e38f;
#pragma unroll
      for (int j = 0; j < 4; ++j) { vv[j] = SC[r * SN + lane + 32 * j]; mx = fmaxf(mx, vv[j]); }
      mx = wmaxr(mx);
      float s = 0.f;
#pragma unroll
      for (int j = 0; j < 4; ++j) { vv[j] = __expf(vv[j] - mx); s += vv[j]; }
      s = wsum(s);
      float inv = 1.f / s;
#pragma unroll
      for (int j = 0; j < 4; ++j) SC[r * SN + lane + 32 * j] = vv[j] * inv;
    }
    __syncthreads();
    { v8f acc1[1] = {};  // ctx_h = att @ V_h  -> overwrite Q columns of head h
      gemm_acc<1, false>(SC, SN, Vbuf + h * HD, DN, SN, acc1, wid, lane);
      store_tiles<1>(Qbuf + h * HD, DN, acc1, wid, lane); }
    __syncthreads();
  }

  // ---- phase 4: O projection + Möbius residual -> X2 (into Vbuf) ----
  rowwise_clip(Qbuf, LCLIP, wid, lane);   // logmap0(expmap0(ctx))
  __syncthreads();
  { v8f acc[8] = {}; gemm_acc<8, true>(Qbuf, DN, Wo, DN, DN, acc, wid, lane);
    store_tiles<8>(Hbuf, DN, acc, wid, lane); }
  __syncthreads();
  for (int r = wid; r < SN; r += WAVES) {
    float z[4]; float sz = 0.f, szy = 0.f;
#pragma unroll
    for (int j = 0; j < 4; ++j) {
      int c = lane + 32 * j;
      z[j] = Hbuf[r * DN + c]; sz += z[j] * z[j]; szy += z[j] * yo[c];
    }
    sz = wsum(sz); szy = wsum(szy);
    float zn = sqrtf(fmaxf(sz, 1e-15f));
    float th = tanhf(zn), s1 = th / zn;
    float xy = s1 * szy, x2 = th * th;
    float y2 = y2s[3];
    float A1 = 1.f + 2.f * xy + y2, B1 = 1.f - x2;
    float den = fmaxf(1.f + 2.f * xy + x2 * y2, 1e-15f);
    float nm = sqrtf(fmaxf(A1 * A1 * x2 + 2.f * A1 * B1 * xy + B1 * B1 * y2, 1e-30f)) / den;
    float np = fminf(nm, MAXN_C);
    float pf = np / nm;                                 // projx only: stay on manifold
    float al = pf * A1 * s1 / den, be = pf * B1 / den;
    float hv[4], xr[4]; float hx = 0.f, r2 = 0.f;
#pragma unroll
    for (int j = 0; j < 4; ++j) {
      int c = lane + 32 * j;
      hv[j] = al * z[j] + be * yo[c];
      xr[j] = xg[r * DN + c];
      hx += hv[j] * xr[j]; r2 += xr[j] * xr[j];
    }
    hx = wsum(hx); r2 = wsum(r2);
    float h2a = np * np;
    float A2 = 1.f + 2.f * hx + r2, B2 = 1.f - h2a;
    float den2 = fmaxf(1.f + 2.f * hx + h2a * r2, 1e-15f);
    float nm2 = sqrtf(fmaxf(A2 * A2 * h2a + 2.f * A2 * B2 * hx + B2 * B2 * r2, 1e-30f)) / den2;
    float ps = fminf(nm2, MAXN_C) / nm2;
#pragma unroll
    for (int j = 0; j < 4; ++j) {
      int c = lane + 32 * j;
      Vbuf[r * DN + c] = ps * (A2 * hv[j] + B2 * xr[j]) / den2;
    }
  }
  __syncthreads();

  // ---- phase 5: H = clip(LN2(logmap0(X2))) ; zero FFN row accumulators ----
  rowwise_ln_clip(Vbuf, DN, l2g + e * DN, l2b + e * DN, Hbuf, wid, lane, LCLIP);
  for (int i = tid; i < SN; i += NTHR) { z1sq[i] = 0.f; z1y[i] = 0.f; rlsq[i] = 0.f; }
  __syncthreads();

  // ---- phase 6: FFN pass 1 — accumulate per-row <z1,z1>, <z1,ytil1> ----
  for (int c = 0; c < 4; ++c) {
    { v8f acc[8] = {};
      gemm_acc<8, true>(Hbuf, DN, F1 + c * 128 * DN, DN, DN, acc, wid, lane);
      store_tiles<8>(Qbuf, DN, acc, wid, lane); }
    __syncthreads();
    for (int r = wid; r < SN; r += WAVES) {
      float ss = 0.f, sy = 0.f;
#pragma unroll
      for (int j = 0; j < 4; ++j) {
        int cc = lane + 32 * j;
        float z = Qbuf[r * DN + cc];
        ss += z * z; sy += z * y1v[c * 128 + cc];
      }
      ss = wsum(ss); sy = wsum(sy);
      if (lane == 0) { z1sq[r] += ss; z1y[r] += sy; }
    }
    __syncthreads();
  }
  // per-row alpha/beta for the fc1 Möbius+ReLU tail
  if (tid < SN) {
    float sz = z1sq[tid];
    float zn = sqrtf(fmaxf(sz, 1e-15f));
    float th = tanhf(zn), s1 = th / zn;
    float xy = s1 * z1y[tid], x2 = th * th;
    float y2 = y2s[5];
    float A1 = 1.f + 2.f * xy + y2, B1 = 1.f - x2;
    float den = fmaxf(1.f + 2.f * xy + x2 * y2, 1e-15f);
    float nm = sqrtf(fmaxf(A1 * A1 * x2 + 2.f * A1 * B1 * xy + B1 * B1 * y2, 1e-30f)) / den;
    float np = fminf(nm, MAXN_C);
    float f = atanhf(np) / nm;
    alA[tid] = f * A1 * s1 / den; beA[tid] = f * B1 / den;
  }
  __syncthreads();

  // ---- phase 7: FFN pass 2 — recompute chunk, ReLU, fc2 WMMA accumulation ----
  v8f acc2[8] = {};
  for (int c = 0; c < 4; ++c) {
    v8f acc[8] = {};
    gemm_acc<8, true>(Hbuf, DN, F1 + c * 128 * DN, DN, DN, acc, wid, lane);
    __syncthreads();                               // prior fc2 reads of Qbuf done
    store_tiles<8>(Qbuf, DN, acc, wid, lane);
    __syncthreads();
    for (int r = wid; r < SN; r += WAVES) {
      float srel = 0.f;
#pragma unroll
      for (int j = 0; j < 4; ++j) {
        int cc = lane + 32 * j;
        float w = alA[r] * Qbuf[r * DN + cc] + beA[r] * y1v[c * 128 + cc];
        float t = fmaxf(w, 0.f);
        Qbuf[r * DN + cc] = t;
        srel += t * t;
      }
      srel = wsum(srel);
      if (lane == 0) rlsq[r] += srel;
    }
    __syncthreads();
    gemm_acc<8, true>(Qbuf, DN, F2 + c * 128, FN, DN, acc2, wid, lane);
  }
  __syncthreads();
  store_tiles<8>(Kbuf, DN, acc2, wid, lane);       // z2
  __syncthreads();

  // ---- phase 8: fc2 Möbius tail + mob_relu + Möbius residual -> out ----
  for (int r = wid; r < SN; r += WAVES) {
    float nr = sqrtf(fmaxf(rlsq[r], 1e-15f));
    float s4 = fminf(nr, LCLIP) / nr;                // logmap0(projx(expmap0(relu)))
    float z[4]; float sz = 0.f, szy = 0.f;
#pragma unroll
    for (int j = 0; j < 4; ++j) {
      int cc = lane + 32 * j;
      z[j] = s4 * Kbuf[r * DN + cc];
      sz += z[j] * z[j]; szy += z[j] * y2v[cc];
    }
    sz = wsum(sz); szy = wsum(szy);
    float zn = sqrtf(fmaxf(sz, 1e-15f));
    float th = tanhf(zn), s1 = th / zn;
    float xy = s1 * szy, x2 = th * th;
    float y2 = y2s[4];
    float A1 = 1.f + 2.f * xy + y2, B1 = 1.f - x2;
    float den = fmaxf(1.f + 2.f * xy + x2 * y2, 1e-15f);
    float nm = sqrtf(fmaxf(A1 * A1 * x2 + 2.f * A1 * B1 * xy + B1 * B1 * y2, 1e-30f)) / den;
    float np = fminf(nm, MAXN_C);
    float f = atanhf(np) / nm;                       // projx + logmap0 of man_linear out
    float al = f * A1 * s1 / den, be = f * B1 / den;
    float t[4]; float st = 0.f;
#pragma unroll
    for (int j = 0; j < 4; ++j) {
      int cc = lane + 32 * j;
      float u = al * z[j] + be * y2v[cc];            // logmap0 domain
      t[j] = fmaxf(u, 0.f);                          // ReLU
      st += t[j] * t[j];
    }
    st = wsum(st);
    float ntr = sqrtf(fmaxf(st, 1e-15f));
    float tht = tanhf(ntr), se = tht / ntr;          // expmap0
    float h2a = tht * tht;
    float hx = 0.f, r2 = 0.f; float xr[4];
#pragma unroll
    for (int j = 0; j < 4; ++j) {
      int cc = lane + 32 * j;
      t[j] *= se;
      xr[j] = Vbuf[r * DN + cc];                     // residual x2
      hx += t[j] * xr[j]; r2 += xr[j] * xr[j];
    }
    hx = wsum(hx); r2 = wsum(r2);
    float A2 = 1.f + 2.f * hx + r2, B2 = 1.f - h2a;
    float den2 = fmaxf(1.f + 2.f * hx + h2a * r2, 1e-15f);
    float nm2 = sqrtf(fmaxf(A2 * A2 * h2a + 2.f * A2 * B2 * hx + B2 * B2 * r2, 1e-30f)) / den2;
    float ps = fminf(nm2, MAXN_C) / nm2;             // final projx
#pragma unroll
    for (int j = 0; j < 4; ++j) {
      int cc = lane + 32 * j;
      out_be[r * DN + cc] = ps * (A2 * t[j] + B2 * xr[j]) / den2;
    }
  }
}

#define LDS_FLOATS (4 * SN * DN + 4 * DN + DN + FN + SN + 5 * SN + 8)
#define LDS_BYTES (LDS_FLOATS * sizeof(float))

extern "C" void kernel_launch(void* const* d_in, const int* in_sizes, int n_in,
                              void* d_out, int out_size, void* d_ws, size_t ws_size,
                              hipStream_t stream) {
  (void)in_sizes; (void)n_in; (void)out_size; (void)d_ws; (void)ws_size;
  const float* feat = (const float*)d_in[0];
  const int* eidx = (const int*)d_in[1];
  const unsigned char* pmask = (const unsigned char*)d_in[2];
  const float* l1g = (const float*)d_in[3];
  const float* l1b = (const float*)d_in[4];
  const float* l2g = (const float*)d_in[5];
  const float* l2b = (const float*)d_in[6];
  const float* pwq = (const float*)d_in[7];
  const float* pbq = (const float*)d_in[8];
  const float* pwk = (const float*)d_in[9];
  const float* pbk = (const float*)d_in[10];
  const float* pwv = (const float*)d_in[11];
  const float* pbv = (const float*)d_in[12];
  const float* pwo = (const float*)d_in[13];
  const float* pbo = (const float*)d_in[14];
  const float* pf1w = (const float*)d_in[15];
  const float* pf1b = (const float*)d_in[16];
  const float* pf2w = (const float*)d_in[17];
  const float* pf2b = (const float*)d_in[18];
  float* out = (float*)d_out;

  (void)hipFuncSetAttribute((const void*)PM_MoE_Att_Block_53824530153848_kernel,
                            hipFuncAttributeMaxDynamicSharedMemorySize, (int)LDS_BYTES);
  dim3 grid(NB * EN), block(NTHR);
  PM_MoE_Att_Block_53824530153848_kernel<<<grid, block, LDS_BYTES, stream>>>(
      feat, eidx, pmask, l1g, l1b, l2g, l2b,
      pwq, pbq, pwk, pbk, pwv, pbv, pwo, pbo,
      pf1w, pf1b, pf2w, pf2b, out);
}